// Propagation_20572893348791
// MI455X (gfx1250) — compile-verified
//
#include <hip/hip_runtime.h>

typedef __attribute__((ext_vector_type(2))) float v2f;
typedef __attribute__((ext_vector_type(8))) float v8f;

#define SCAN_B 256
#define ALPHA_C 0.1f
#define DF 50            // feature dim per reference

// ---------------- degree count ----------------
__global__ void k_count(const int* __restrict__ dst, int E, int* __restrict__ deg) {
    int i = blockIdx.x * blockDim.x + threadIdx.x;
    if (i < E) atomicAdd(&deg[dst[i]], 1);
}

// ---------------- block-local exclusive scan ----------------
__global__ void k_scan1(const int* __restrict__ deg, int N,
                        int* __restrict__ rp, int* __restrict__ bsum) {
    __shared__ int sh[SCAN_B];
    int t = threadIdx.x;
    int i = blockIdx.x * SCAN_B + t;
    int v = (i < N) ? deg[i] : 0;
    sh[t] = v;
    __syncthreads();
    for (int off = 1; off < SCAN_B; off <<= 1) {
        int add = (t >= off) ? sh[t - off] : 0;
        __syncthreads();
        sh[t] += add;
        __syncthreads();
    }
    if (i < N) rp[i] = sh[t] - v;
    if (t == SCAN_B - 1) bsum[blockIdx.x] = sh[t];
}

// ---------------- scan of block sums (nb <= 1024) ----------------
__global__ void k_scan2(int* __restrict__ bsum, int nb) {
    __shared__ int sh[1024];
    int t = threadIdx.x;
    int v = (t < nb) ? bsum[t] : 0;
    sh[t] = v;
    __syncthreads();
    for (int off = 1; off < 1024; off <<= 1) {
        int add = (t >= off) ? sh[t - off] : 0;
        __syncthreads();
        sh[t] += add;
        __syncthreads();
    }
    if (t < nb) bsum[t] = sh[t] - v;
}

// ---------------- finalize row_ptr, cursor, norm ----------------
__global__ void k_finish(const int* __restrict__ deg, int N, int E,
                         int* __restrict__ rp, const int* __restrict__ bsum,
                         int* __restrict__ cur, float* __restrict__ nrm) {
    int i = blockIdx.x * SCAN_B + threadIdx.x;
    if (i < N) {
        int r = rp[i] + bsum[blockIdx.x];
        rp[i] = r;
        cur[i] = r;
        float d = fmaxf((float)deg[i], 1e-12f);
        float x = rsqrtf(d);
        x = x * (1.5f - 0.5f * d * x * x);        // Newton refine
        nrm[i] = x;
    }
    if (i == 0) rp[N] = E;
}

// ---------------- scatter edges into CSR ----------------
__global__ void k_scatter(const int* __restrict__ src, const int* __restrict__ dst, int E,
                          int* __restrict__ cur, int* __restrict__ col) {
    int e = blockIdx.x * blockDim.x + threadIdx.x;
    if (e < E) {
        int p = atomicAdd(&cur[dst[e]], 1);
        col[p] = src[e];
    }
}

// ---------------- one 4-edge WMMA window (fully branch-free) ----------------
// A(16x4): A[m][k] = norm[src_k] iff edge k owned by row m. Ownership encoded as
// a {0,1} float multiplied onto the loaded norm: under strict FP the compiler
// cannot sink the norm load under the predicate (w*0 != 0 for NaN), so the
// gather chain stays unconditional -> no exec-mask branches in the hot loop.
template<int SIN>
__device__ __forceinline__ void do_window(int w, int gEnd, int rpm, int rpm1,
                                          int hi, int m,
                                          const float* __restrict__ hin,
                                          const float* __restrict__ nrm,
                                          const int* __restrict__ col,
                                          v8f acc[4]) {
    int ea = w + (hi << 1);
    int eb = ea + 1;
    int eac = min(ea, gEnd - 1);            // always valid inside the edge loop
    int ebc = min(eb, gEnd - 1);
    int sA = col[eac];
    int sB = col[ebc];
    float wA = nrm[sA];
    float wB = nrm[sB];
    float m0 = (ea >= rpm && ea < rpm1) ? 1.0f : 0.0f;
    float m1 = (eb >= rpm && eb < rpm1) ? 1.0f : 0.0f;
    v2f Aop = {wA * m0, wB * m1};
    int oA = sA * SIN;                      // 32-bit index math -> scale_offset addr
    int oB = sB * SIN;
#pragma unroll
    for (int t = 0; t < 4; ++t) {
        int c = t * 16 + m;
        if (SIN < 64) { if (c >= SIN) c = 0; }   // pad cols never touch cols < 50
        v2f Bop = {hin[oA + c], hin[oB + c]};
        acc[t] = __builtin_amdgcn_wmma_f32_16x16x4_f32(
            false, Aop, false, Bop, (short)0, acc[t], false, false);
    }
}

// ---------------- WMMA propagation step ----------------
template<int SIN, int SOUT>
__global__ __launch_bounds__(256)
void k_prop(const float* __restrict__ hin,
            float* __restrict__ hout,
            const float* __restrict__ feat0,
            const float* __restrict__ nrm,
            const int* __restrict__ rp,
            const int* __restrict__ col,
            int N) {
    int wid  = (blockIdx.x * 256 + threadIdx.x) >> 5;   // wave id = 16-row group
    int lane = threadIdx.x & 31;
    int nGroups = (N + 15) >> 4;
    if (wid >= nGroups) return;                          // wave-uniform

    int gbase = wid << 4;
    int m  = lane & 15;
    int hi = lane >> 4;
    int row_m = gbase + m;
    int rpm, rpm1;
    if (row_m < N) { rpm = rp[row_m]; rpm1 = rp[row_m + 1]; }
    else           { rpm = 1; rpm1 = 0; }
    int gStart = rp[gbase];
    int gEnd   = rp[min(gbase + 16, N)];

    // two accumulator sets; 4 windows (16 edges) in flight per iteration:
    // occupancy is launch-limited (~6 waves/SIMD), so per-wave MLP hides L2 latency
    v8f accP[4], accQ[4];
#pragma unroll
    for (int t = 0; t < 4; ++t) { accP[t] = v8f{}; accQ[t] = v8f{}; }

    for (int w = gStart; w < gEnd; w += 16) {
        do_window<SIN>(w,      gEnd, rpm, rpm1, hi, m, hin, nrm, col, accP);
        do_window<SIN>(w + 4,  gEnd, rpm, rpm1, hi, m, hin, nrm, col, accQ);
        do_window<SIN>(w + 8,  gEnd, rpm, rpm1, hi, m, hin, nrm, col, accP);
        do_window<SIN>(w + 12, gEnd, rpm, rpm1, hi, m, hin, nrm, col, accQ);
    }

    // D layout: VGPR r -> row r (lanes 0-15) / r+8 (lanes 16-31), col = lane%16
    const float one_m_a = 1.0f - ALPHA_C;
#pragma unroll
    for (int r = 0; r < 8; ++r) {
        int row = gbase + r + (hi << 3);
        if (row >= N) continue;
        float scale = one_m_a * nrm[row];
#pragma unroll
        for (int t = 0; t < 4; ++t) {
            int c = t * 16 + m;
            float val = (accP[t][r] + accQ[t][r]) * scale;
            if (c < DF)   val += ALPHA_C * feat0[(size_t)row * DF + c];
            if (c < SOUT) hout[(size_t)row * SOUT + c] = val;
        }
    }
}

extern "C" void kernel_launch(void* const* d_in, const int* in_sizes, int n_in,
                              void* d_out, int out_size, void* d_ws, size_t ws_size,
                              hipStream_t stream) {
    const float* feat = (const float*)d_in[0];
    const int*   eidx = (const int*)d_in[1];
    const int E  = in_sizes[1] / 2;
    const int N  = in_sizes[0] / DF;
    const int* src = eidx;                   // edge_index row 0
    const int* dst = eidx + E;               // edge_index row 1

    // workspace carve-out (256B aligned): ~34 MB total
    char* w = (char*)d_ws;
    auto carve = [&](size_t bytes) -> char* {
        char* p = w; w += (bytes + 255) & ~(size_t)255; return p;
    };
    float* Abuf = (float*)carve((size_t)N * 64 * sizeof(float)); // padded ping buffer
    int*   deg  = (int*)  carve((size_t)N * sizeof(int));
    int*   rp   = (int*)  carve((size_t)(N + 1) * sizeof(int));
    int*   cur  = (int*)  carve((size_t)N * sizeof(int));
    int*   col  = (int*)  carve((size_t)E * sizeof(int));
    int nb = (N + SCAN_B - 1) / SCAN_B;      // 391 for N=100000 (<=1024 required)
    int*   bsum = (int*)  carve((size_t)nb * sizeof(int));
    float* nrm  = (float*)carve((size_t)N * sizeof(float));

    // ---- CSR build (rebuilt every call) ----
    hipMemsetAsync(deg, 0, (size_t)N * sizeof(int), stream);
    k_count  <<<(E + 255) / 256, 256, 0, stream>>>(dst, E, deg);
    k_scan1  <<<nb, SCAN_B, 0, stream>>>(deg, N, rp, bsum);
    k_scan2  <<<1, 1024, 0, stream>>>(bsum, nb);
    k_finish <<<nb, SCAN_B, 0, stream>>>(deg, N, E, rp, bsum, cur, nrm);
    k_scatter<<<(E + 255) / 256, 256, 0, stream>>>(src, dst, E, cur, col);

    // ---- 10 propagation steps: odd -> Abuf (stride 64), even -> d_out (stride 50)
    int nGroups = (N + 15) / 16;
    int blocks  = (nGroups * 32 + 255) / 256;
    for (int s = 1; s <= 10; ++s) {
        if (s & 1) {
            const float* in = (s == 1) ? feat : (const float*)d_out;
            k_prop<DF, 64><<<blocks, 256, 0, stream>>>(in, Abuf, feat, nrm, rp, col, N);
        } else {
            k_prop<64, DF><<<blocks, 256, 0, stream>>>(Abuf, (float*)d_out, feat, nrm, rp, col, N);
        }
    }
}